// WolfPQMinDistEncoder_78520592106002
// MI455X (gfx1250) — compile-verified
//
#include <hip/hip_runtime.h>
#include <hip/hip_bf16.h>
#include <math.h>
#include <stdint.h>

// Problem constants (match reference)
#define NN   8192
#define DIMC 768
#define MC   64
#define KC   256
#define SUB  12
#define H1C  8192            // M*K/2
#define MKC  16384           // M*K

typedef float v2f __attribute__((ext_vector_type(2)));
typedef float v8f __attribute__((ext_vector_type(8)));

#define LDA1 34   // gemm1 LDS pitch
#define LDA2 36   // gemm2 A pitch: 144B rows -> 16B-aligned b128 async writes
#define LDB2 34   // gemm2 B^T pitch: conflict-free b32 scatter, 8B-aligned b64 frags

// ---------------------------------------------------------------------------
// Kernel 1: LayerNorm  xn = (x - mu)*rsqrt(var+eps)*gamma + beta   (one row/block)
// ---------------------------------------------------------------------------
__global__ __launch_bounds__(256) void ln_kernel(const float* __restrict__ x,
                                                 const float* __restrict__ g,
                                                 const float* __restrict__ b,
                                                 float* __restrict__ xn) {
  __shared__ float red[256];
  const int n = blockIdx.x, tid = threadIdx.x;
  const float v0 = x[n * DIMC + tid];
  const float v1 = x[n * DIMC + 256 + tid];
  const float v2 = x[n * DIMC + 512 + tid];
  red[tid] = v0 + v1 + v2;
  __syncthreads();
  for (int s = 128; s > 0; s >>= 1) {
    if (tid < s) red[tid] += red[tid + s];
    __syncthreads();
  }
  const float mu = red[0] * (1.0f / DIMC);
  __syncthreads();
  const float d0 = v0 - mu, d1 = v1 - mu, d2 = v2 - mu;
  red[tid] = d0 * d0 + d1 * d1 + d2 * d2;
  __syncthreads();
  for (int s = 128; s > 0; s >>= 1) {
    if (tid < s) red[tid] += red[tid + s];
    __syncthreads();
  }
  const float rstd = rsqrtf(red[0] * (1.0f / DIMC) + 1e-5f);
  xn[n * DIMC + tid]       = d0 * rstd * g[tid]       + b[tid];
  xn[n * DIMC + 256 + tid] = d1 * rstd * g[256 + tid] + b[256 + tid];
  xn[n * DIMC + 512 + tid] = d2 * rstd * g[512 + tid] + b[512 + tid];
}

// ---------------------------------------------------------------------------
// Kernel 2: PQ argmin index per (n, m):  kmin[n][m] = argmin_k ||x_m - c_mk||^2
// ---------------------------------------------------------------------------
__global__ __launch_bounds__(256) void pq_argmin_kernel(const float* __restrict__ x,
                                                        const float* __restrict__ cb,
                                                        int* __restrict__ kmin) {
  __shared__ float xs[DIMC];
  __shared__ float sd[256];
  __shared__ int   si[256];
  const int n = blockIdx.x, tid = threadIdx.x;
  for (int i = tid; i < DIMC; i += 256) xs[i] = x[n * DIMC + i];
  __syncthreads();
  for (int m = 0; m < MC; ++m) {
    const float* c = cb + ((size_t)(m * KC + tid)) * SUB;
    float d = 0.f;
#pragma unroll
    for (int q = 0; q < SUB; ++q) {
      const float t = xs[m * SUB + q] - c[q];
      d += t * t;
    }
    sd[tid] = d; si[tid] = tid;
    __syncthreads();
    for (int s = 128; s > 0; s >>= 1) {
      if (tid < s) {
        const float dv = sd[tid + s]; const int iv = si[tid + s];
        if (dv < sd[tid] || (dv == sd[tid] && iv < si[tid])) { sd[tid] = dv; si[tid] = iv; }
      }
      __syncthreads();
    }
    if (tid == 0) kmin[n * MC + m] = si[0];
    __syncthreads();
  }
}

// ---------------------------------------------------------------------------
// Kernel 3: GEMM1 + tanh:  h = tanh(xn @ W1 + b1)   (8192x768x8192, fp32 WMMA)
// ---------------------------------------------------------------------------
__global__ __launch_bounds__(256) void gemm1_tanh_kernel(const float* __restrict__ xn,
                                                         const float* __restrict__ W1,
                                                         const float* __restrict__ b1,
                                                         float* __restrict__ h) {
  __shared__ float As[128][LDA1];
  __shared__ float Bts[64][LDA1];
  const int n0 = blockIdx.y * 128;
  const int c0 = blockIdx.x * 64;
  const int tid = threadIdx.x;
  const int lane = tid & 31, w = tid >> 5;
  const int wr = w & 3, wc = w >> 2;      // 4 row groups x 2 col groups
  const int l16 = lane & 15, lh = lane >> 4;

  v8f acc[2][2] = {};
  for (int k0 = 0; k0 < DIMC; k0 += 32) {
#pragma unroll
    for (int i = 0; i < 16; ++i) {        // A: 128x32
      const int li = i * 256 + tid;
      const int r = li >> 5, kk = li & 31;
      As[r][kk] = xn[(size_t)(n0 + r) * DIMC + k0 + kk];
    }
#pragma unroll
    for (int i = 0; i < 8; ++i) {         // B^T: 32x64 -> [c][k]
      const int li = i * 256 + tid;
      const int kk = li >> 6, c = li & 63;
      Bts[c][kk] = W1[(size_t)(k0 + kk) * H1C + c0 + c];
    }
    __syncthreads();
#pragma unroll
    for (int kk = 0; kk < 32; kk += 4) {
      const int ko = kk + 2 * lh;
      v2f a[2], bfr[2];
      a[0]   = *(const v2f*)&As[wr * 32 + l16][ko];
      a[1]   = *(const v2f*)&As[wr * 32 + 16 + l16][ko];
      bfr[0] = *(const v2f*)&Bts[wc * 32 + l16][ko];
      bfr[1] = *(const v2f*)&Bts[wc * 32 + 16 + l16][ko];
#pragma unroll
      for (int i = 0; i < 2; ++i)
#pragma unroll
        for (int j = 0; j < 2; ++j)
          acc[i][j] = __builtin_amdgcn_wmma_f32_16x16x4_f32(
              false, a[i], false, bfr[j], (short)0, acc[i][j], false, false);
    }
    __syncthreads();
  }
#pragma unroll
  for (int i = 0; i < 2; ++i)
#pragma unroll
    for (int j = 0; j < 2; ++j) {
      const int col = c0 + wc * 32 + j * 16 + l16;
      const float bias = b1[col];
#pragma unroll
      for (int e = 0; e < 8; ++e) {
        const int row = n0 + wr * 32 + i * 16 + e + 8 * lh;
        h[(size_t)row * H1C + col] = tanhf(acc[i][j][e] + bias);
      }
    }
}

// ---------------------------------------------------------------------------
// Kernel 4: GEMM2 fused, double-buffered with CDNA5 async global->LDS copies.
// Block tile 128 rows x 256 cols (full K segment of one m). 8 waves 2x4,
// each wave 64x64 (4x4 fp32 WMMA tiles). ASYNCcnt-tracked staging overlaps
// next-stage HBM/L2 traffic with the current 128-WMMA compute phase.
// ---------------------------------------------------------------------------
__global__ __launch_bounds__(256) void gemm2_fused_kernel(const float* __restrict__ h,
                                                          const float* __restrict__ W2,
                                                          const float* __restrict__ b2,
                                                          const float* __restrict__ interp_param,
                                                          const int*   __restrict__ kmin,
                                                          const float* __restrict__ gumbel,
                                                          float* __restrict__ out) {
  __shared__ float As[2][128][LDA2];
  __shared__ float Bts[2][256][LDB2];
  __shared__ float pval[128][4];
  __shared__ int   pidx[128][4];
  __shared__ int   amax[128];

  const int m  = blockIdx.x;          // 0..63
  const int n0 = blockIdx.y * 128;
  const int tid = threadIdx.x, lane = tid & 31, w = tid >> 5;
  const int wr = w & 1, wc = w >> 1;  // 2 row groups x 4 col groups
  const int l16 = lane & 15, lh = lane >> 4;
  const int colbase = m * KC;

  // --- async staging addresses -------------------------------------------
  // A tile (128x32 f32) via b128: issue i -> row = i*32 + tid/8, 16B chunk (tid&7)
  const uint64_t gA0 = (uint64_t)(uintptr_t)(h + (size_t)(n0 + (tid >> 3)) * H1C + (tid & 7) * 4);
  const unsigned lA0 = (unsigned)(uintptr_t)&As[0][tid >> 3][(tid & 7) * 4];
  // B tile (32x256 f32, transposed scatter) via b32: issue i -> kk = i, col = tid
  const uint64_t gB0 = (uint64_t)(uintptr_t)(W2 + (size_t)colbase + tid);
  const unsigned lB0 = (unsigned)(uintptr_t)&Bts[0][tid][0];
  const unsigned ABUF = 128 * LDA2 * 4;   // bytes per A buffer
  const unsigned BBUF = 256 * LDB2 * 4;   // bytes per B buffer

  auto issue_stage = [&](int s, int buf) {
    const uint64_t ga = gA0 + (uint64_t)s * 128;                 // +32 floats in K
    const unsigned la = lA0 + (unsigned)buf * ABUF;
#pragma unroll
    for (int i = 0; i < 4; ++i) {
      const unsigned ld = la + (unsigned)(i * 32 * LDA2 * 4);
      const uint64_t gm = ga + (uint64_t)i * (32u * H1C * 4u);
      asm volatile("global_load_async_to_lds_b128 %0, %1, off"
                   :: "v"(ld), "v"(gm) : "memory");
    }
    const uint64_t gb = gB0 + (uint64_t)s * (32ull * MKC * 4ull);
    const unsigned lb = lB0 + (unsigned)buf * BBUF;
#pragma unroll
    for (int i = 0; i < 32; ++i) {
      const unsigned ld = lb + (unsigned)(i * 4);
      const uint64_t gm = gb + (uint64_t)i * ((uint64_t)MKC * 4ull);
      asm volatile("global_load_async_to_lds_b32 %0, %1, off"
                   :: "v"(ld), "v"(gm) : "memory");
    }
  };

  v8f acc[4][4] = {};
  issue_stage(0, 0);
  for (int s = 0; s < H1C / 32; ++s) {
    const int buf = s & 1;
    asm volatile("s_wait_asynccnt 0x0" ::: "memory");  // this wave's copies done
    __syncthreads();                                   // => all waves' copies done
    if (s + 1 < H1C / 32) issue_stage(s + 1, buf ^ 1); // overlap with compute
#pragma unroll
    for (int kk = 0; kk < 32; kk += 4) {
      const int ko = kk + 2 * lh;
      v2f a[4], bfr[4];
#pragma unroll
      for (int i = 0; i < 4; ++i) a[i]   = *(const v2f*)&As[buf][wr * 64 + i * 16 + l16][ko];
#pragma unroll
      for (int j = 0; j < 4; ++j) bfr[j] = *(const v2f*)&Bts[buf][wc * 64 + j * 16 + l16][ko];
#pragma unroll
      for (int i = 0; i < 4; ++i)
#pragma unroll
        for (int j = 0; j < 4; ++j)
          acc[i][j] = __builtin_amdgcn_wmma_f32_16x16x4_f32(
              false, a[i], false, bfr[j], (short)0, acc[i][j], false, false);
    }
    __syncthreads();  // all waves done reading buf before it is refilled
  }

  // ---- fused epilogue: bias+relu+interp-combine+gumbel -> argmax -> one-hot ----
  const float ip = 1.0f / (1.0f + expf(-interp_param[0]));   // sigmoid
  const float om = 1.0f - ip;
  const float boost = ip * 100.0f;                           // MIN_DIST_FACTOR

#pragma unroll
  for (int i = 0; i < 4; ++i) {
#pragma unroll
    for (int e = 0; e < 8; ++e) {
      const int rloc = wr * 64 + i * 16 + e + 8 * lh;        // 0..127
      const int row  = n0 + rloc;
      const int km   = kmin[row * MC + m];
      const size_t gbase = ((size_t)row * MC + m) * KC;
      float bestv = -1e30f; int bestc = 0;
#pragma unroll
      for (int j = 0; j < 4; ++j) {
        const int col = wc * 64 + j * 16 + l16;              // 0..255
        float a = acc[i][j][e] + b2[colbase + col];
        a = a > 0.f ? a : 0.f;                               // relu
        const float lg = om * a + (col == km ? boost : 0.0f) + gumbel[gbase + col];
        if (lg > bestv) { bestv = lg; bestc = col; }
      }
#pragma unroll
      for (int s = 1; s < 16; s <<= 1) {                     // half-wave reduce
        const float ov = __shfl_xor(bestv, s, 32);
        const int   oc = __shfl_xor(bestc, s, 32);
        if (ov > bestv || (ov == bestv && oc < bestc)) { bestv = ov; bestc = oc; }
      }
      if (l16 == 0) { pval[rloc][wc] = bestv; pidx[rloc][wc] = bestc; }
    }
  }
  __syncthreads();
  if (tid < 128) {
    float bv = pval[tid][0]; int bc = pidx[tid][0];
#pragma unroll
    for (int q = 1; q < 4; ++q) {
      const float v = pval[tid][q]; const int c2 = pidx[tid][q];
      if (v > bv || (v == bv && c2 < bc)) { bv = v; bc = c2; }
    }
    amax[tid] = bc;
  }
  __syncthreads();
  for (int r = 0; r < 128; ++r) {      // one-hot write, fully coalesced
    const int best = amax[r];
    out[((size_t)(n0 + r) * MC + m) * KC + tid] = (tid == best) ? 1.0f : 0.0f;
  }
}

// ---------------------------------------------------------------------------
extern "C" void kernel_launch(void* const* d_in, const int* in_sizes, int n_in,
                              void* d_out, int out_size, void* d_ws, size_t ws_size,
                              hipStream_t stream) {
  const float* x      = (const float*)d_in[0];
  const float* cb     = (const float*)d_in[1];
  const float* lng    = (const float*)d_in[2];
  const float* lnb    = (const float*)d_in[3];
  const float* W1     = (const float*)d_in[4];
  const float* b1     = (const float*)d_in[5];
  const float* W2     = (const float*)d_in[6];
  const float* b2     = (const float*)d_in[7];
  const float* interp = (const float*)d_in[8];
  const float* gum    = (const float*)d_in[9];
  float* out = (float*)d_out;

  float* xn   = (float*)d_ws;                       // N*DIM fp32   (~25 MB)
  float* h    = xn + (size_t)NN * DIMC;             // N*H1 fp32    (256 MB)
  int*   kmin = (int*)(h + (size_t)NN * H1C);       // N*M  int32   (2 MB)

  ln_kernel<<<NN, 256, 0, stream>>>(x, lng, lnb, xn);
  pq_argmin_kernel<<<NN, 256, 0, stream>>>(x, cb, kmin);
  gemm1_tanh_kernel<<<dim3(H1C / 64, NN / 128), 256, 0, stream>>>(xn, W1, b1, h);
  gemm2_fused_kernel<<<dim3(MC, NN / 128), 256, 0, stream>>>(h, W2, b2, interp, kmin, gum, out);
}